// SHINE_13331578487561
// MI455X (gfx1250) — compile-verified
//
#include <hip/hip_runtime.h>
#include <hip/hip_bf16.h>
#include <math.h>

typedef __attribute__((ext_vector_type(2))) float v2f;
typedef __attribute__((ext_vector_type(8))) float v8f;

// ---------------------------------------------------------------------------
// zero: grid-stride fill
// ---------------------------------------------------------------------------
__global__ void zero_f32(float* __restrict__ p, int n) {
    int stride = gridDim.x * blockDim.x;
    for (int i = blockIdx.x * blockDim.x + threadIdx.x; i < n; i += stride)
        p[i] = 0.0f;
}

// ---------------------------------------------------------------------------
// relu: grid-stride elementwise
// ---------------------------------------------------------------------------
__global__ void relu_f32(float* __restrict__ p, int n) {
    int stride = gridDim.x * blockDim.x;
    for (int i = blockIdx.x * blockDim.x + threadIdx.x; i < n; i += stride) {
        float v = p[i];
        p[i] = v > 0.0f ? v : 0.0f;
    }
}

// ---------------------------------------------------------------------------
// SpMM scatter: y[rows[e], coloff + f] += vals[e] * x[cols[e], f]
// One wave32 per edge; lanes stride the feature dimension (d<=428 -> <=14
// atomics per lane). Random rows => atomic scatter is the right path; the
// whole edge+feature working set lives in the 192MB L2.
// ---------------------------------------------------------------------------
__global__ void spmm_scatter(const int* __restrict__ rows,
                             const int* __restrict__ cols,
                             const float* __restrict__ vals, int nnz,
                             const float* __restrict__ x, int dx,
                             float* __restrict__ y, int ldy, int coloff, int d) {
    int lane = threadIdx.x & 31;
    int wid  = (blockIdx.x * blockDim.x + threadIdx.x) >> 5;
    int nw   = (gridDim.x * blockDim.x) >> 5;
    for (int e = wid; e < nnz; e += nw) {
        int r = rows[e];
        int c = cols[e];
        float v = vals[e];
        const float* xs = x + (size_t)c * dx;
        float* yd = y + (size_t)r * ldy + coloff;
        for (int f = lane; f < d; f += 32)
            atomicAdd(&yd[f], v * xs[f]);
    }
}

// ---------------------------------------------------------------------------
// Dense transform Y = X @ W + bias, X:[M,128], W:[128,128], M % 16 == 0.
// One wave per 16x16 output tile; K swept in steps of 4 with
// v_wmma_f32_16x16x4_f32 (f32 in/out: no precision downgrade vs reference).
//
// Fragment layouts (ISA 7.12.2):
//   A (16x4 f32, 2 VGPRs): lanes 0-15 -> M=lane,  K = k+0 / k+1
//                          lanes 16-31 -> M=lane-16, K = k+2 / k+3
//   B (4x16 f32, 2 VGPRs): mirrored -> lanes 0-15 K=k+0/k+1, lanes16-31 K=k+2/3
//   C/D (16x16 f32, 8 VGPRs): VGPR r -> M = r + 8*(lane/16), N = lane%16
// ---------------------------------------------------------------------------
__global__ void gemm128_wmma(const float* __restrict__ X,
                             const float* __restrict__ W,
                             const float* __restrict__ bias,
                             float* __restrict__ Y, int M) {
    int wave = (blockIdx.x * blockDim.x + threadIdx.x) >> 5;
    int lane = threadIdx.x & 31;
    int tiles = (M >> 4) * 8;           // (M/16) x (128/16)
    if (wave >= tiles) return;          // uniform per wave; EXEC stays all-1s

    int tm = wave >> 3;                 // tile row
    int tn = wave & 7;                  // tile col
    int half = lane >> 4;               // 0: lanes 0-15, 1: lanes 16-31
    int l16  = lane & 15;

    const float* xrow = X + (size_t)(tm * 16 + l16) * 128;
    const float* wcol = W + tn * 16 + l16;

    v8f c = {};
    #pragma unroll
    for (int k = 0; k < 128; k += 4) {
        int ka = k + 2 * half;
        v2f a, b;
        a[0] = xrow[ka];
        a[1] = xrow[ka + 1];
        b[0] = wcol[(size_t)ka * 128];
        b[1] = wcol[(size_t)(ka + 1) * 128];
        c = __builtin_amdgcn_wmma_f32_16x16x4_f32(
                /*neg_a=*/false, a, /*neg_b=*/false, b,
                /*c_mod=*/(short)0, c, /*reuse_a=*/false, /*reuse_b=*/false);
    }

    float bn = bias[tn * 16 + l16];
    #pragma unroll
    for (int r = 0; r < 8; ++r) {
        int m = tm * 16 + r + 8 * half;
        Y[(size_t)m * 128 + tn * 16 + l16] = c[r] + bn;
    }
}

// ---------------------------------------------------------------------------
// Naive GEMM for the tiny POS-branch transforms (M=60): Y = X @ W + bias
// ---------------------------------------------------------------------------
__global__ void gemm_naive(const float* __restrict__ X, const float* __restrict__ W,
                           const float* __restrict__ bias, float* __restrict__ Y,
                           int M, int K, int N) {
    int idx = blockIdx.x * blockDim.x + threadIdx.x;
    if (idx >= M * N) return;
    int m = idx / N, n = idx % N;
    float acc = bias[n];
    for (int k = 0; k < K; ++k)
        acc += X[m * K + k] * W[k * N + n];
    Y[idx] = acc;
}

// ---------------------------------------------------------------------------
// Row-wise L2 normalization: p[row,:] /= (||p[row,:]||_2 + 1e-9)
// One block (256 threads) per row, LDS tree reduction.
// ---------------------------------------------------------------------------
__global__ void l2norm_rows(float* __restrict__ y, int ld) {
    __shared__ float red[256];
    float* p = y + (size_t)blockIdx.x * ld;
    float s = 0.0f;
    for (int f = threadIdx.x; f < ld; f += blockDim.x) {
        float v = p[f];
        s += v * v;
    }
    red[threadIdx.x] = s;
    __syncthreads();
    for (int off = blockDim.x >> 1; off > 0; off >>= 1) {
        if ((int)threadIdx.x < off) red[threadIdx.x] += red[threadIdx.x + off];
        __syncthreads();
    }
    float inv = 1.0f / (sqrtf(red[0]) + 1e-9f);
    for (int f = threadIdx.x; f < ld; f += blockDim.x)
        p[f] *= inv;
}

// ---------------------------------------------------------------------------
// Host orchestration
// ---------------------------------------------------------------------------
extern "C" void kernel_launch(void* const* d_in, const int* in_sizes, int n_in,
                              void* d_out, int out_size, void* d_ws, size_t ws_size,
                              hipStream_t stream) {
    (void)in_sizes; (void)n_in; (void)out_size; (void)ws_size;

    const int N_DOC = 10000, N_WORD = 20000, N_ENT = 10000, N_POS = 60;
    const int D = 128, D_WEMB = 300;
    const int E11 = 320000, E22 = 160000, E33 = 3600;
    const int E01 = 300000, E02 = 100000, E03 = 150000;

    const float* f1       = (const float*)d_in[0];
    const float* f2       = (const float*)d_in[1];
    const float* f3       = (const float*)d_in[2];
    const float* word_emb = (const float*)d_in[3];
    const int*   a11r = (const int*)d_in[4];
    const int*   a11c = (const int*)d_in[5];
    const float* a11v = (const float*)d_in[6];
    const int*   a22r = (const int*)d_in[7];
    const int*   a22c = (const int*)d_in[8];
    const float* a22v = (const float*)d_in[9];
    const int*   a33r = (const int*)d_in[10];
    const int*   a33c = (const int*)d_in[11];
    const float* a33v = (const float*)d_in[12];
    const int*   a01r = (const int*)d_in[13];
    const int*   a01c = (const int*)d_in[14];
    const float* a01v = (const float*)d_in[15];
    const int*   a02r = (const int*)d_in[16];
    const int*   a02c = (const int*)d_in[17];
    const float* a02v = (const float*)d_in[18];
    const int*   a03r = (const int*)d_in[19];
    const int*   a03c = (const int*)d_in[20];
    const float* a03v = (const float*)d_in[21];
    const float* W3   = (const float*)d_in[22];
    const float* b3   = (const float*)d_in[23];
    const float* W1_2 = (const float*)d_in[24];
    const float* b1_2 = (const float*)d_in[25];
    const float* W2_2 = (const float*)d_in[26];
    const float* b2_2 = (const float*)d_in[27];
    const float* W3_2 = (const float*)d_in[28];
    const float* b3_2 = (const float*)d_in[29];

    float* out  = (float*)d_out;
    float* out1 = out;                                   // [10000,128]
    float* out2 = out + (size_t)N_DOC * D;               // [10000,428]
    float* out3 = out2 + (size_t)N_DOC * (D + D_WEMB);   // [10000,128]

    float* bufA = (float*)d_ws;                          // 20000*128
    float* bufB = bufA + (size_t)N_WORD * D;             // 20000*128
    float* bufC = bufB + (size_t)N_WORD * D;             // 60*128
    float* bufD = bufC + (size_t)N_POS * D;              // 60*128

    const int TB = 256;
    auto zgrid = [](int n) { int b = (n + 255) / 256; return b > 16384 ? 16384 : b; };
    auto egrid = [](int nnz) { int b = (nnz + 7) / 8; return b > 65535 ? 65535 : b; };

    // ---- type 1: words [20000,128] ----
    zero_f32<<<zgrid(N_WORD * D), TB, 0, stream>>>(bufA, N_WORD * D);
    spmm_scatter<<<egrid(E11), TB, 0, stream>>>(a11r, a11c, a11v, E11, f1, D, bufA, D, 0, D);
    relu_f32<<<zgrid(N_WORD * D), TB, 0, stream>>>(bufA, N_WORD * D);
    gemm128_wmma<<<(N_WORD / 16) * 8 / 4, 128, 0, stream>>>(bufA, W1_2, b1_2, bufB, N_WORD);
    zero_f32<<<zgrid(N_WORD * D), TB, 0, stream>>>(bufA, N_WORD * D);
    spmm_scatter<<<egrid(E11), TB, 0, stream>>>(a11r, a11c, a11v, E11, bufB, D, bufA, D, 0, D);
    relu_f32<<<zgrid(N_WORD * D), TB, 0, stream>>>(bufA, N_WORD * D);
    zero_f32<<<zgrid(N_DOC * D), TB, 0, stream>>>(out1, N_DOC * D);
    spmm_scatter<<<egrid(E01), TB, 0, stream>>>(a01r, a01c, a01v, E01, bufA, D, out1, D, 0, D);
    l2norm_rows<<<N_DOC, 256, 0, stream>>>(out1, D);

    // ---- type 2: entities [10000,128] (+ word_emb concat via column offset) ----
    zero_f32<<<zgrid(N_ENT * D), TB, 0, stream>>>(bufA, N_ENT * D);
    spmm_scatter<<<egrid(E22), TB, 0, stream>>>(a22r, a22c, a22v, E22, f2, D, bufA, D, 0, D);
    relu_f32<<<zgrid(N_ENT * D), TB, 0, stream>>>(bufA, N_ENT * D);
    gemm128_wmma<<<(N_ENT / 16) * 8 / 4, 128, 0, stream>>>(bufA, W2_2, b2_2, bufB, N_ENT);
    zero_f32<<<zgrid(N_ENT * D), TB, 0, stream>>>(bufA, N_ENT * D);
    spmm_scatter<<<egrid(E22), TB, 0, stream>>>(a22r, a22c, a22v, E22, bufB, D, bufA, D, 0, D);
    relu_f32<<<zgrid(N_ENT * D), TB, 0, stream>>>(bufA, N_ENT * D);
    zero_f32<<<zgrid(N_DOC * (D + D_WEMB)), TB, 0, stream>>>(out2, N_DOC * (D + D_WEMB));
    spmm_scatter<<<egrid(E02), TB, 0, stream>>>(a02r, a02c, a02v, E02, bufA, D,
                                                out2, D + D_WEMB, 0, D);
    spmm_scatter<<<egrid(E02), TB, 0, stream>>>(a02r, a02c, a02v, E02, word_emb, D_WEMB,
                                                out2, D + D_WEMB, D, D_WEMB);
    l2norm_rows<<<N_DOC, 256, 0, stream>>>(out2, D + D_WEMB);

    // ---- type 3: POS [60,128] ----
    gemm_naive<<<(N_POS * D + TB - 1) / TB, TB, 0, stream>>>(f3, W3, b3, bufC, N_POS, 60, D);
    zero_f32<<<zgrid(N_POS * D), TB, 0, stream>>>(bufD, N_POS * D);
    spmm_scatter<<<egrid(E33), TB, 0, stream>>>(a33r, a33c, a33v, E33, bufC, D, bufD, D, 0, D);
    relu_f32<<<zgrid(N_POS * D), TB, 0, stream>>>(bufD, N_POS * D);
    gemm_naive<<<(N_POS * D + TB - 1) / TB, TB, 0, stream>>>(bufD, W3_2, b3_2, bufC, N_POS, D, D);
    zero_f32<<<zgrid(N_POS * D), TB, 0, stream>>>(bufD, N_POS * D);
    spmm_scatter<<<egrid(E33), TB, 0, stream>>>(a33r, a33c, a33v, E33, bufC, D, bufD, D, 0, D);
    relu_f32<<<zgrid(N_POS * D), TB, 0, stream>>>(bufD, N_POS * D);
    zero_f32<<<zgrid(N_DOC * D), TB, 0, stream>>>(out3, N_DOC * D);
    spmm_scatter<<<egrid(E03), TB, 0, stream>>>(a03r, a03c, a03v, E03, bufD, D, out3, D, 0, D);
    l2norm_rows<<<N_DOC, 256, 0, stream>>>(out3, D);
}